// EmbeddingShardV2_45131516347064
// MI455X (gfx1250) — compile-verified
//
#include <hip/hip_runtime.h>

// Embedding gather: out[row, :] = W[x[row], :] + b,  row in [0, B*S)
// B=4, S=2048, D_MODEL=1024, VOCAB=50304, all float32.
//
// Pure bandwidth problem (~64 MB moved, ~2.7 us at 23.3 TB/s). WMMA is
// deliberately NOT used: the one-hot matmul formulation would be ~8.5e11 MACs
// and >200 MB of mandatory W traffic — orders of magnitude off the gather's
// bandwidth bound. CDNA5 features used instead: B128 coalesced accesses,
// non-temporal (TH=NT) output stores to preserve L2 for gathered W rows, and
// global_prefetch_b8 of the next gathered row to hide random-access latency.

typedef float v4f __attribute__((ext_vector_type(4)));

#ifndef D_MODEL
#define D_MODEL 1024
#endif

__global__ __launch_bounds__(256) void embed_gather_kernel(
    const int* __restrict__ x,    // [nrows] token ids
    const float* __restrict__ W,  // [VOCAB, D_MODEL]
    const float* __restrict__ b,  // [D_MODEL]
    float* __restrict__ out,      // [nrows, D_MODEL]
    int nrows)
{
    const int t = threadIdx.x;            // 0..255, one float4 column per lane
    // Bias chunk for this lane's column: loaded once, reused for every row
    // this block processes (bias is 4 KB; L2-resident after first touch).
    const v4f bias = ((const v4f*)b)[t];

    for (int row = blockIdx.x; row < nrows; row += gridDim.x) {
        const int tok = x[row];  // uniform per block -> scalar load/broadcast

        // Prefetch this lane's 16B chunk of the NEXT row this block will
        // gather (random row -> likely HBM; emits global_prefetch_b8).
        const int nrow = row + gridDim.x;
        if (nrow < nrows) {
            const int ntok = x[nrow];
            __builtin_prefetch(
                (const void*)(W + (size_t)ntok * D_MODEL + (size_t)t * 4),
                /*rw=*/0, /*locality=*/1);
        }

        // Gather: 256 lanes x float4 = 1024 floats = one full row.
        // global_load_b128, fully coalesced (512B per wave).
        const v4f* __restrict__ src = (const v4f*)(W + (size_t)tok * D_MODEL);
        v4f v = src[t];
        v += bias;

        // Non-temporal global_store_b128: output is never re-read on device;
        // keep it from evicting gathered W rows out of the 192 MB L2.
        v4f* __restrict__ dst = (v4f*)(out + (size_t)row * D_MODEL);
        __builtin_nontemporal_store(v, dst + t);
    }
}

extern "C" void kernel_launch(void* const* d_in, const int* in_sizes, int n_in,
                              void* d_out, int out_size, void* d_ws, size_t ws_size,
                              hipStream_t stream) {
    const int*   x = (const int*)  d_in[0];  // [B*S] int32
    const float* W = (const float*)d_in[1];  // [VOCAB, D_MODEL] f32
    const float* b = (const float*)d_in[2];  // [D_MODEL] f32
    float*     out = (float*)d_out;          // [B*S, D_MODEL] f32

    const int nrows = in_sizes[0];           // 8192

    // 2048 blocks x 8 wave32 = 16384 waves in flight; each block walks 4 rows
    // so the next-row prefetch has a target. 256 threads = 256 float4 = 1 row.
    const int blocks = 2048;
    embed_gather_kernel<<<dim3(blocks), dim3(256), 0, stream>>>(x, W, b, out, nrows);
}